// FraudGTStyle_9483287790245
// MI455X (gfx1250) — compile-verified
//
#include <hip/hip_runtime.h>
#include <hip/hip_bf16.h>
#include <math.h>

#define N_NODES 50000
#define N_EDGES 800000
#define HID 128
#define HEADS 4
#define LAYERS 2
#define EDGE_DIM 4
#define LDSTRIDE 132   // pad 128-float rows to avoid LDS bank conflicts (64 banks)

typedef __attribute__((ext_vector_type(2))) float v2f;
typedef __attribute__((ext_vector_type(8))) float v8f;

// ---------------------------------------------------------------------------
// Y[M,128] = act( X[M,128] @ W[128,128] + bias )   (ACT: 0=none, 1=relu)
// grid.x = M/16 (M is a multiple of 16), block = 256 threads = 8 wave32.
// Each block stages its 16x128 X strip in LDS; wave w computes the 16x16
// output tile at columns [16w, 16w+16) with 32 chained V_WMMA_F32_16X16X4_F32.
// ---------------------------------------------------------------------------
template <int ACT>
__global__ __launch_bounds__(256) void gemm128_wmma(
    const float* __restrict__ X, const float* __restrict__ W,
    const float* __restrict__ bias, float* __restrict__ Y)
{
    __shared__ float lds[16 * LDSTRIDE];
    const int m0  = blockIdx.x * 16;
    const int tid = threadIdx.x;

    // cooperative, coalesced load of the 16x128 A strip
    for (int i = tid; i < 16 * HID; i += 256) {
        int r = i >> 7, c = i & 127;
        lds[r * LDSTRIDE + c] = X[(size_t)(m0 + r) * HID + c];
    }
    __syncthreads();

    const int wave = tid >> 5;
    const int lane = tid & 31;
    const int n0   = wave * 16;
    const int half = lane >> 4;   // 0: K pair {k,k+1}, 1: K pair {k+2,k+3}
    const int l16  = lane & 15;

    v8f acc = {};
#pragma unroll 4
    for (int k = 0; k < HID; k += 4) {
        const int kk = k + half * 2;
        v2f a, b;
        // A (16x4, MxK): lane l16 holds row M=l16, K = kk, kk+1
        a.x = lds[l16 * LDSTRIDE + kk];
        a.y = lds[l16 * LDSTRIDE + kk + 1];
        // B (4x16, KxN): lane l16 holds col N=n0+l16, K = kk, kk+1
        b.x = W[(size_t)kk * HID + n0 + l16];
        b.y = W[(size_t)(kk + 1) * HID + n0 + l16];
        acc = __builtin_amdgcn_wmma_f32_16x16x4_f32(
            false, a, false, b, (short)0, acc, false, false);
    }

    const float bn = bias[n0 + l16];
#pragma unroll
    for (int g = 0; g < 8; ++g) {
        const int m = m0 + g + half * 8;   // C/D layout: VGPR g -> M = g (+8 for hi half)
        float v = acc[g] + bn;
        if (ACT == 1) v = v > 0.f ? v : 0.f;
        Y[(size_t)m * HID + n0 + l16] = v;
    }
}

// ---------------------------------------------------------------------------
// Segment-softmax state init: m = -inf, denom = 0   (n = N_NODES*HEADS)
// ---------------------------------------------------------------------------
__global__ void init_softmax_state(float* __restrict__ m, float* __restrict__ denom, int n)
{
    int i = blockIdx.x * blockDim.x + threadIdx.x;
    if (i < n) { m[i] = -__builtin_inff(); denom[i] = 0.f; }
}

// ---------------------------------------------------------------------------
// attn[e,h] = (q[dst[e],h,:] . k[src[e],h,:]) / sqrt(32) + edge_attr[e] @ Wb[:,h] + bb[h]
// plus float segment-max into mbuf[dst,h] via the signed/unsigned atomic trick.
// One thread per (edge, head).
// ---------------------------------------------------------------------------
__global__ __launch_bounds__(256) void attn_kernel(
    const float* __restrict__ q, const float* __restrict__ k,
    const int* __restrict__ src, const int* __restrict__ dst,
    const float* __restrict__ edge_attr,
    const float* __restrict__ Wb, const float* __restrict__ bb,
    float* __restrict__ attn, float* __restrict__ mbuf)
{
    int t = blockIdx.x * 256 + threadIdx.x;
    if (t >= N_EDGES * HEADS) return;
    const int e  = t >> 2;
    const int hh = t & 3;
    const int s = src[e], d = dst[e];

    const float4* qp = (const float4*)(q + (size_t)d * HID + hh * 32);
    const float4* kp = (const float4*)(k + (size_t)s * HID + hh * 32);
    float acc = 0.f;
#pragma unroll
    for (int i = 0; i < 8; ++i) {
        float4 a = qp[i], b = kp[i];
        acc += a.x * b.x + a.y * b.y + a.z * b.z + a.w * b.w;
    }

    const float* ea = edge_attr + (size_t)e * EDGE_DIM;
    float eb = bb[hh];
#pragma unroll
    for (int j = 0; j < EDGE_DIM; ++j) eb += ea[j] * Wb[j * HEADS + hh];

    const float val = acc * 0.17677669529663687f + eb;   // 1/sqrt(32)
    attn[t] = val;

    float* addr = mbuf + (size_t)d * HEADS + hh;
    if (val >= 0.f) atomicMax((int*)addr, __float_as_int(val));
    else            atomicMin((unsigned int*)addr, (unsigned int)__float_as_int(val));
}

// ---------------------------------------------------------------------------
// attn[e,h] = exp(attn[e,h] - m[dst,h]);  denom[dst,h] += attn[e,h]
// ---------------------------------------------------------------------------
__global__ __launch_bounds__(256) void exp_kernel(
    const int* __restrict__ dst, float* __restrict__ attn,
    const float* __restrict__ mbuf, float* __restrict__ denom)
{
    int t = blockIdx.x * 256 + threadIdx.x;
    if (t >= N_EDGES * HEADS) return;
    const int e  = t >> 2;
    const int hh = t & 3;
    const int d = dst[e];
    const float val = expf(attn[t] - mbuf[(size_t)d * HEADS + hh]);
    attn[t] = val;
    atomicAdd(denom + (size_t)d * HEADS + hh, val);
}

// ---------------------------------------------------------------------------
// aggr[dst,c] += (attn[e,h]/denom[dst,h]) * v[src,c] * sigmoid(ea[e] @ Wg[:,c] + bg[c])
// 128 threads per edge -> coalesced v/aggr access.
// ---------------------------------------------------------------------------
__global__ __launch_bounds__(256) void message_kernel(
    const float* __restrict__ v, const int* __restrict__ src, const int* __restrict__ dst,
    const float* __restrict__ attn, const float* __restrict__ denom,
    const float* __restrict__ edge_attr,
    const float* __restrict__ Wg, const float* __restrict__ bg,
    float* __restrict__ aggr)
{
    int t = blockIdx.x * 256 + threadIdx.x;
    const int e = t >> 7;
    const int c = t & 127;
    if (e >= N_EDGES) return;
    const int s = src[e], d = dst[e];
    const int hh = c >> 5;

    const float alpha = attn[(size_t)e * HEADS + hh] /
                        (denom[(size_t)d * HEADS + hh] + 1e-16f);

    const float* ea = edge_attr + (size_t)e * EDGE_DIM;
    float g = bg[c];
#pragma unroll
    for (int j = 0; j < EDGE_DIM; ++j) g += ea[j] * Wg[j * HID + c];
    g = 1.f / (1.f + expf(-g));

    const float msg = alpha * v[(size_t)s * HID + c] * g;
    atomicAdd(aggr + (size_t)d * HID + c, msg);
}

// ---------------------------------------------------------------------------
// h = relu(aggr + h)   (aggr already holds h@Ws + bs + scatter-adds)
// ---------------------------------------------------------------------------
__global__ void combine_relu(float* __restrict__ h, const float* __restrict__ aggr, int n)
{
    int i = blockIdx.x * blockDim.x + threadIdx.x;
    if (i < n) {
        float v = aggr[i] + h[i];
        h[i] = v > 0.f ? v : 0.f;
    }
}

// ---------------------------------------------------------------------------
// out[n,o] = h[n,:] @ W_out[:,o] + b_out[o]   (OUT_CH = 2)
// ---------------------------------------------------------------------------
__global__ void out_proj(const float* __restrict__ h, const float* __restrict__ W,
                         const float* __restrict__ b, float* __restrict__ out)
{
    int t = blockIdx.x * blockDim.x + threadIdx.x;
    if (t >= N_NODES * 2) return;
    const int n = t >> 1, o = t & 1;
    const float4* hp = (const float4*)(h + (size_t)n * HID);
    float acc = b[o];
#pragma unroll 8
    for (int i = 0; i < 32; ++i) {
        float4 hv = hp[i];
        acc += hv.x * W[(i * 4 + 0) * 2 + o] + hv.y * W[(i * 4 + 1) * 2 + o] +
               hv.z * W[(i * 4 + 2) * 2 + o] + hv.w * W[(i * 4 + 3) * 2 + o];
    }
    out[t] = acc;
}

// ---------------------------------------------------------------------------
extern "C" void kernel_launch(void* const* d_in, const int* in_sizes, int n_in,
                              void* d_out, int out_size, void* d_ws, size_t ws_size,
                              hipStream_t stream)
{
    (void)in_sizes; (void)n_in; (void)out_size; (void)ws_size;

    const float* x         = (const float*)d_in[0];
    const int*   ei        = (const int*)  d_in[1];
    const float* edge_attr = (const float*)d_in[2];
    const float* W_in      = (const float*)d_in[3];
    const float* b_in      = (const float*)d_in[4];
    const float* Wq        = (const float*)d_in[5];
    const float* bq        = (const float*)d_in[6];
    const float* Wk        = (const float*)d_in[7];
    const float* bk        = (const float*)d_in[8];
    const float* Wv        = (const float*)d_in[9];
    const float* bv        = (const float*)d_in[10];
    const float* Wb        = (const float*)d_in[11];
    const float* bb        = (const float*)d_in[12];
    const float* Wg        = (const float*)d_in[13];
    const float* bg        = (const float*)d_in[14];
    const float* Ws        = (const float*)d_in[15];
    const float* bs        = (const float*)d_in[16];
    const float* W_out     = (const float*)d_in[17];
    const float* b_out     = (const float*)d_in[18];

    const int* src = ei;             // edge_index[0]
    const int* dst = ei + N_EDGES;   // edge_index[1]

    // workspace carve (floats): ~143 MB total
    float* ws    = (float*)d_ws;
    const size_t NH = (size_t)N_NODES * HID;
    float* h     = ws;
    float* q     = h    + NH;
    float* kbuf  = q    + NH;
    float* vbuf  = kbuf + NH;
    float* aggr  = vbuf + NH;
    float* attn  = aggr + NH;
    float* mbuf  = attn + (size_t)N_EDGES * HEADS;
    float* denom = mbuf + (size_t)N_NODES * HEADS;

    const dim3 blk(256);
    const int gemm_grid = N_NODES / 16;                         // 3125 (exact)
    const int eh_grid   = (N_EDGES * HEADS + 255) / 256;
    const int msg_grid  = (int)(((size_t)N_EDGES * HID + 255) / 256);
    const int nh_grid   = (int)((NH + 255) / 256);
    const int nm_grid   = (N_NODES * HEADS + 255) / 256;

    // h = relu(x @ W_in + b_in)
    gemm128_wmma<1><<<gemm_grid, blk, 0, stream>>>(x, W_in, b_in, h);

    for (int l = 0; l < LAYERS; ++l) {
        const float* Wql = Wq + (size_t)l * HID * HID;  const float* bql = bq + (size_t)l * HID;
        const float* Wkl = Wk + (size_t)l * HID * HID;  const float* bkl = bk + (size_t)l * HID;
        const float* Wvl = Wv + (size_t)l * HID * HID;  const float* bvl = bv + (size_t)l * HID;
        const float* Wsl = Ws + (size_t)l * HID * HID;  const float* bsl = bs + (size_t)l * HID;
        const float* Wbl = Wb + (size_t)l * EDGE_DIM * HEADS; const float* bbl = bb + (size_t)l * HEADS;
        const float* Wgl = Wg + (size_t)l * EDGE_DIM * HID;   const float* bgl = bg + (size_t)l * HID;

        gemm128_wmma<0><<<gemm_grid, blk, 0, stream>>>(h, Wql, bql, q);
        gemm128_wmma<0><<<gemm_grid, blk, 0, stream>>>(h, Wkl, bkl, kbuf);
        gemm128_wmma<0><<<gemm_grid, blk, 0, stream>>>(h, Wvl, bvl, vbuf);
        // skip path written straight into the aggregation buffer
        gemm128_wmma<0><<<gemm_grid, blk, 0, stream>>>(h, Wsl, bsl, aggr);

        init_softmax_state<<<nm_grid, blk, 0, stream>>>(mbuf, denom, N_NODES * HEADS);
        attn_kernel<<<eh_grid, blk, 0, stream>>>(q, kbuf, src, dst, edge_attr,
                                                 Wbl, bbl, attn, mbuf);
        exp_kernel<<<eh_grid, blk, 0, stream>>>(dst, attn, mbuf, denom);
        message_kernel<<<msg_grid, blk, 0, stream>>>(vbuf, src, dst, attn, denom,
                                                     edge_attr, Wgl, bgl, aggr);
        combine_relu<<<nh_grid, blk, 0, stream>>>(h, aggr, (int)NH);
    }

    out_proj<<<(N_NODES * 2 + 255) / 256, blk, 0, stream>>>(h, W_out, b_out, (float*)d_out);
}